// GAandFilter_50388556316750
// MI455X (gfx1250) — compile-verified
//
#include <hip/hip_runtime.h>

// ---------------- problem constants (match reference) ----------------------
constexpr int   BN   = 8;
constexpr int   HH   = 608;
constexpr int   WW   = 968;
constexpr int   NPIX = HH * WW;          // per-batch pixels
constexpr int   NT   = BN * NPIX;        // total pixels
constexpr float LOG2E  = 1.44269504088896340736f;
constexpr float INV2SS = 1.0f / (2.0f * 3.0f * 3.0f);   // sigma_s = 3
constexpr float INV2SR = 1.0f / (2.0f * 0.1f * 0.1f);   // sigma_r = 0.1
constexpr float K2R    = -INV2SR * LOG2E;               // range-kernel log2 slope

#define PB 128   // partial blocks per batch for the reduction

// ---------------- CDNA5 async global->LDS path (guarded) -------------------
#if defined(__gfx1250__) && __has_builtin(__builtin_amdgcn_global_load_async_to_lds_b32)
#define HAVE_ASYNC 1
#else
#define HAVE_ASYNC 0
#endif

#if HAVE_ASYNC
typedef __attribute__((address_space(1))) int gint_t;   // prints as __device__ int*
typedef __attribute__((address_space(3))) int lint_t;   // prints as __shared__ int*

__device__ __forceinline__ void async_g2l_b32(const float* gsrc, float* ldst) {
  // global source: flat address == as1 address on AMDGPU; integer round-trip
  // also drops const. LDS dest: explicit addrspacecast generic -> as3.
  gint_t* gp = (gint_t*)(unsigned long long)gsrc;
  lint_t* lp = (lint_t*)(int*)ldst;
  __builtin_amdgcn_global_load_async_to_lds_b32(gp, lp, 0, 0);
}
__device__ __forceinline__ void wait_async0() {
#if __has_builtin(__builtin_amdgcn_s_wait_asynccnt)
  __builtin_amdgcn_s_wait_asynccnt(0);
#else
  asm volatile("s_wait_asynccnt 0" ::: "memory");
#endif
}
#endif

// ---------------- K1: per-batch partial sums (deterministic) ---------------
__global__ __launch_bounds__(256)
void k_reduce(const float* __restrict__ est, const float* __restrict__ pd,
              double* __restrict__ partials) {
  __shared__ double s0[256], s1[256], s2[256], s3[256], s4[256];
  const int t  = threadIdx.x;
  const int b  = blockIdx.y;
  const int N4 = NPIX / 4;                     // 147136, base 16B-aligned
  const float4* e4 = (const float4*)(est + b * NPIX);
  const float4* p4 = (const float4*)(pd  + b * NPIX);
  const int stride = gridDim.x * blockDim.x;   // PB*256
  double a00 = 0.0, a01 = 0.0, a11 = 0.0, b0 = 0.0, b1 = 0.0;
  for (int i = blockIdx.x * blockDim.x + t; i < N4; i += stride) {
    if (i + stride < N4) {
      __builtin_prefetch(&e4[i + stride], 0, 1);   // global_prefetch_b8
      __builtin_prefetch(&p4[i + stride], 0, 1);
    }
    const float4 e = e4[i];
    const float4 p = p4[i];
    const float ev[4] = {e.x, e.y, e.z, e.w};
    const float pv[4] = {p.x, p.y, p.z, p.w};
#pragma unroll
    for (int c = 0; c < 4; ++c) {
      if (pv[c] > 0.1f && pv[c] < 12.0f) {
        const float tg = 1.0f / pv[c];       // f32 target, like the reference
        a00 += (double)(ev[c] * ev[c]);
        a01 += (double)ev[c];
        a11 += 1.0;
        b0  += (double)(ev[c] * tg);
        b1  += (double)tg;
      }
    }
  }
  s0[t] = a00; s1[t] = a01; s2[t] = a11; s3[t] = b0; s4[t] = b1;
  __syncthreads();
  for (int s = 128; s > 0; s >>= 1) {
    if (t < s) {
      s0[t] += s0[t + s]; s1[t] += s1[t + s]; s2[t] += s2[t + s];
      s3[t] += s3[t + s]; s4[t] += s4[t + s];
    }
    __syncthreads();
  }
  if (t == 0) {
    double* o = partials + (size_t)(b * gridDim.x + blockIdx.x) * 5;
    o[0] = s0[0]; o[1] = s1[0]; o[2] = s2[0]; o[3] = s3[0]; o[4] = s4[0];
  }
}

// ---------------- K2: finalize scale/shift per batch -----------------------
__global__ __launch_bounds__(PB)
void k_finalize(const double* __restrict__ partials, float* __restrict__ hdr) {
  __shared__ double s0[PB], s1[PB], s2[PB], s3[PB], s4[PB];
  const int t = threadIdx.x;
  const int b = blockIdx.x;
  const double* p = partials + (size_t)(b * PB + t) * 5;
  s0[t] = p[0]; s1[t] = p[1]; s2[t] = p[2]; s3[t] = p[3]; s4[t] = p[4];
  __syncthreads();
  for (int s = PB / 2; s > 0; s >>= 1) {
    if (t < s) {
      s0[t] += s0[t + s]; s1[t] += s1[t + s]; s2[t] += s2[t + s];
      s3[t] += s3[t + s]; s4[t] += s4[t + s];
    }
    __syncthreads();
  }
  if (t == 0) {
    const double a00 = s0[0], a01 = s1[0], a11 = s2[0], b0 = s3[0], b1 = s4[0];
    const double det = a00 * a11 - a01 * a01;
    const double sf  = ( a11 * b0 - a01 * b1) / (det + 1e-12);
    const double hf  = (-a01 * b0 + a00 * b1) / (det + 1e-12);
    float scale, shift;
    if (sf < 0.0) { scale = (float)(b0 / (a00 + 1e-12)); shift = 0.0f; }
    else          { scale = (float)sf;                   shift = (float)hf; }
    hdr[2 * b]     = scale;
    hdr[2 * b + 1] = shift;
  }
}

// ---------------- K3: joint bilateral (9x9), tiled in LDS ------------------
// 32x16 output tile + 4-pixel halo -> 40x24 LDS tile of (guide, residual)
// packed as float2 so each tap is a single conflict-free ds_load_b64.
__global__ __launch_bounds__(512)
void k_filter(const float* __restrict__ est, const float* __restrict__ pd,
              const float* __restrict__ hdr, float* __restrict__ out) {
  constexpr int TX = 32, TY = 16, IW = TX + 8, IH = TY + 8, NLDS = IW * IH;
  __shared__ float2 sGR[IH][IW];
  const int   b     = blockIdx.z;
  const float scale = hdr[2 * b];
  const float shift = hdr[2 * b + 1];
  const int x0 = blockIdx.x * TX - 4;
  const int y0 = blockIdx.y * TY - 4;
  const int tx = threadIdx.x, ty = threadIdx.y;
  const int tid  = ty * TX + tx;
  const int base = b * NPIX;

#if HAVE_ASYNC
  // Stage raw est/prompt halo tiles into LDS with async global->LDS loads,
  // then compute (ga, residual) from LDS.
  __shared__ float stE[NLDS];
  __shared__ float stP[NLDS];
  for (int idx = tid; idx < NLDS; idx += TX * TY) {
    const int ly = idx / IW, lx = idx % IW;
    const int gy = min(max(y0 + ly, 0), HH - 1);
    const int gx = min(max(x0 + lx, 0), WW - 1);
    const int a  = base + gy * WW + gx;
    async_g2l_b32(est + a, &stE[idx]);
    async_g2l_b32(pd  + a, &stP[idx]);
  }
  wait_async0();
  __syncthreads();
  for (int idx = tid; idx < NLDS; idx += TX * TY) {
    const int ly = idx / IW, lx = idx % IW;
    const float e  = stE[idx];
    const float p  = stP[idx];
    const float ga = fmaf(e, scale, shift);            // ga_output
    const bool  m  = (p > 0.1f) && (p < 12.0f) && (e > 0.0f);
    sGR[ly][lx] = make_float2(ga, m ? (1.0f / p) / ga : 1.0f);
  }
#else
  // Fallback: direct global loads (edge-clamped == jnp.pad(mode='edge')).
  for (int idx = tid; idx < NLDS; idx += TX * TY) {
    const int ly = idx / IW, lx = idx % IW;
    const int gy = min(max(y0 + ly, 0), HH - 1);
    const int gx = min(max(x0 + lx, 0), WW - 1);
    const int a  = base + gy * WW + gx;
    const float e  = est[a];
    const float p  = pd [a];
    const float ga = fmaf(e, scale, shift);
    const bool  m  = (p > 0.1f) && (p < 12.0f) && (e > 0.0f);
    sGR[ly][lx] = make_float2(ga, m ? (1.0f / p) / ga : 1.0f);
  }
#endif
  __syncthreads();

  const float2 ctr = sGR[ty + 4][tx + 4];
  const float  gc  = ctr.x;
  float num = ctr.y;          // center tap: diff=0, w=1 exactly (ref: gs=1)
  float den = 1.0f;
#pragma unroll
  for (int j = 0; j < 9; ++j) {
#pragma unroll
    for (int i = 0; i < 9; ++i) {
      if (j == 4 && i == 4) continue;    // center handled above
      // spatial gaussian folded into a compile-time log2-domain constant
      const float cs = -(float)((j - 4) * (j - 4) + (i - 4) * (i - 4)) *
                       (INV2SS * LOG2E);
      const float2 gr = sGR[ty + j][tx + i];           // one ds_load_b64
      const float d = gr.x - gc;
      const float w = exp2f(fmaf(d * d, K2R, cs));     // one v_exp_f32
      num = fmaf(w, gr.y, num);
      den += w;
    }
  }

  const int gx = x0 + 4 + tx, gy = y0 + 4 + ty;
  if (gx < WW && gy < HH) {
    const int o = base + gy * WW + gx;
    out[o]          = num / (den + 1e-12f);  // dense_residual
    out[NT + o]     = gc;                    // ga_output
    out[2 * NT + o] = gc;                    // d (== ga_output)
  }
}

// ---------------- host launch ----------------------------------------------
extern "C" void kernel_launch(void* const* d_in, const int* in_sizes, int n_in,
                              void* d_out, int out_size, void* d_ws, size_t ws_size,
                              hipStream_t stream) {
  (void)in_sizes; (void)n_in; (void)out_size; (void)ws_size;
  const float* est = (const float*)d_in[0];   // estimate
  const float* pd  = (const float*)d_in[1];   // prompt_depth
  float* out = (float*)d_out;

  // workspace: [0..64) floats = per-batch (scale, shift); doubles at +256B
  float*  hdr      = (float*)d_ws;
  double* partials = (double*)((char*)d_ws + 256);   // BN*PB*5 doubles = 40KB

  k_reduce  <<<dim3(PB, BN), 256, 0, stream>>>(est, pd, partials);
  k_finalize<<<BN, PB, 0, stream>>>(partials, hdr);

  dim3 grid((WW + 31) / 32, (HH + 15) / 16, BN);
  k_filter<<<grid, dim3(32, 16), 0, stream>>>(est, pd, hdr, out);
}